// ContextQueryAttention_10995116278440
// MI455X (gfx1250) — compile-verified
//
#include <hip/hip_runtime.h>

// Problem constants (from reference): B=64, D=128, N=1024, M=256
#define Bsz 64
#define Dd  128
#define Nn  1024
#define Mm  256

typedef __attribute__((ext_vector_type(2))) float v2f;
typedef __attribute__((ext_vector_type(8))) float v8f;

// D = A(16x4 f32) * B(4x16 f32) + C(16x16 f32)
__device__ __forceinline__ v8f wmma4(v2f a, v2f b, v8f c) {
    return __builtin_amdgcn_wmma_f32_16x16x4_f32(
        /*neg_a=*/false, a, /*neg_b=*/false, b,
        /*c_mod=*/(short)0, c, /*reuse_a=*/false, /*reuse_b=*/false);
}

// ---------------------------------------------------------------------------
// rowterm[b,n] = sum_d C[b,d,n] * w_c[d]   (w_c = W0[D..2D))
__global__ __launch_bounds__(256) void rowterm_kernel(
    const float* __restrict__ C, const float* __restrict__ W0,
    float* __restrict__ rterm) {
    const int b = blockIdx.x;
    const int n = blockIdx.y * 256 + threadIdx.x;
    const float* Cb = C + (long)b * Dd * Nn;
    float acc = 0.0f;
    for (int d = 0; d < Dd; ++d)
        acc += Cb[(long)d * Nn + n] * W0[Dd + d];
    rterm[b * Nn + n] = acc;
}

// colterm[b,m] = sum_d Q[b,d,m] * w_q[d]   (w_q = W0[0..D))
__global__ __launch_bounds__(256) void colterm_kernel(
    const float* __restrict__ Q, const float* __restrict__ W0,
    float* __restrict__ cterm) {
    const int b = blockIdx.x;
    const int m = threadIdx.x;
    const float* Qb = Q + (long)b * Dd * Mm;
    float acc = 0.0f;
    for (int d = 0; d < Dd; ++d)
        acc += Qb[(long)d * Mm + m] * W0[d];
    cterm[b * Mm + m] = acc;
}

// ---------------------------------------------------------------------------
// S[b,n,m] = sum_d (Ct[n,d]*w_qc[d]) * Qt[m,d] + rterm[n] + cterm[m] + b0
// grid (B, N/128), block 256 = 8 waves.
// Wave = (wr = wave>>1, wc = wave&1): TWO 16-row strips at nb, nb+16 sharing
// one B-operand stream over a 128-col half -> 16 WMMA tiles, 128 acc VGPRs.
__global__ __launch_bounds__(256) void s_gemm_kernel(
    const float* __restrict__ C, const float* __restrict__ Q,
    const float* __restrict__ W0, const float* __restrict__ b0,
    const float* __restrict__ rterm, const float* __restrict__ cterm,
    float* __restrict__ S) {
    const int b     = blockIdx.x;
    const int wave  = threadIdx.x >> 5;
    const int lane  = threadIdx.x & 31;
    const int row   = lane & 15;
    const int half  = lane >> 4;
    const int khalf = half << 1;
    const int nb    = blockIdx.y * 128 + (wave >> 1) * 32;
    const int mb    = (wave & 1) * 128;

    const float* Cb  = C + (long)b * Dd * Nn;
    const float* Qb  = Q + (long)b * Dd * Mm;
    const float* Wqc = W0 + 2 * Dd;

    v8f acc0[8] = {};
    v8f acc1[8] = {};

    for (int d0 = 0; d0 < Dd; d0 += 4) {
        const int dA = d0 + khalf;
        const float w0v = Wqc[dA];
        const float w1v = Wqc[dA + 1];
        const float* c0 = Cb + (long)dA * Nn + nb + row;
        v2f a0, a1;                       // A-op: Chat[n, dA..dA+1]
        a0.x = c0[0]       * w0v;
        a0.y = c0[Nn]      * w1v;
        a1.x = c0[16]      * w0v;
        a1.y = c0[Nn + 16] * w1v;
        if (d0 + 8 < Dd)
            __builtin_prefetch(c0 + 8L * Nn, 0, 1);
        #pragma unroll
        for (int mt = 0; mt < 8; ++mt) {
            const int m = mb + mt * 16 + row;
            v2f bv;                       // B-op: Qt[m, dA..dA+1] shared by both strips
            bv.x = Qb[(long)dA * Mm + m];
            bv.y = Qb[(long)(dA + 1) * Mm + m];
            acc0[mt] = wmma4(a0, bv, acc0[mt]);
            acc1[mt] = wmma4(a1, bv, acc1[mt]);
        }
    }

    const float bias = b0[0];
    #pragma unroll
    for (int mt = 0; mt < 8; ++mt) {
        const int m = mb + mt * 16 + row;
        const float ct = cterm[b * Mm + m] + bias;
        #pragma unroll
        for (int r = 0; r < 8; ++r) {
            const int n0 = nb + r + half * 8;   // C/D layout: VGPR r = rows r, r+8
            const int n1 = n0 + 16;
            S[((long)b * Nn + n0) * Mm + m] = acc0[mt][r] + rterm[b * Nn + n0] + ct;
            S[((long)b * Nn + n1) * Mm + m] = acc1[mt][r] + rterm[b * Nn + n1] + ct;
        }
    }
}

// ---------------------------------------------------------------------------
// Row softmax stats: one wave per row (M=256 -> 8 values/lane), shuffle-reduce.
// grid (B, N/8), block 256 = 8 waves.
__global__ __launch_bounds__(256) void rowstats_kernel(
    const float* __restrict__ S, float* __restrict__ rmax,
    float* __restrict__ rsum) {
    const int b    = blockIdx.x;
    const int wave = threadIdx.x >> 5;
    const int lane = threadIdx.x & 31;
    const int n    = blockIdx.y * 8 + wave;
    const float* rowp = S + ((long)b * Nn + n) * Mm;

    float v[8];
    float mx = -3.402823466e38f;
    #pragma unroll
    for (int j = 0; j < 8; ++j) { v[j] = rowp[lane + 32 * j]; mx = fmaxf(mx, v[j]); }
    #pragma unroll
    for (int off = 16; off >= 1; off >>= 1) mx = fmaxf(mx, __shfl_xor(mx, off, 32));
    float sm = 0.0f;
    #pragma unroll
    for (int j = 0; j < 8; ++j) sm += __expf(v[j] - mx);
    #pragma unroll
    for (int off = 16; off >= 1; off >>= 1) sm += __shfl_xor(sm, off, 32);
    if (lane == 0) { rmax[b * Nn + n] = mx; rsum[b * Nn + n] = sm; }
}

// Column softmax stats, phase 1: partial (max, sum) over 128-row chunks.
// grid (B, N/128), block 256 (== M); loads coalesced across m.
__global__ __launch_bounds__(256) void colstats1_kernel(
    const float* __restrict__ S, float* __restrict__ pmax,
    float* __restrict__ psum) {
    const int b  = blockIdx.x;
    const int ch = blockIdx.y;                 // 0..7
    const int m  = threadIdx.x;
    const float* colp = S + ((long)b * Nn + ch * 128) * Mm + m;
    float mx = -3.402823466e38f, sm = 0.0f;
    for (int n = 0; n < 128; ++n) {
        const float vv = colp[(long)n * Mm];
        const float nm = fmaxf(mx, vv);
        sm = sm * __expf(mx - nm) + __expf(vv - nm);
        mx = nm;
    }
    pmax[((long)b * 8 + ch) * Mm + m] = mx;
    psum[((long)b * 8 + ch) * Mm + m] = sm;
}

// Column softmax stats, phase 2: merge 8 partials per column.
// grid (B), block 256 (== M).
__global__ __launch_bounds__(256) void colstats2_kernel(
    const float* __restrict__ pmax, const float* __restrict__ psum,
    float* __restrict__ cmax, float* __restrict__ csum) {
    const int b = blockIdx.x;
    const int m = threadIdx.x;
    float mx = -3.402823466e38f;
    #pragma unroll
    for (int ch = 0; ch < 8; ++ch)
        mx = fmaxf(mx, pmax[((long)b * 8 + ch) * Mm + m]);
    float sm = 0.0f;
    #pragma unroll
    for (int ch = 0; ch < 8; ++ch)
        sm += psum[((long)b * 8 + ch) * Mm + m] *
              __expf(pmax[((long)b * 8 + ch) * Mm + m] - mx);
    cmax[b * Mm + m] = mx;
    csum[b * Mm + m] = sm;
}

// ---------------------------------------------------------------------------
// A[b,n,d] = sum_m softmax_row(S)[n,m] * Qt[m,d]
// grid (B, N/128), block 256; wave: two 16-row strips x 64 cols, K=M=256.
__global__ __launch_bounds__(256) void a_gemm_kernel(
    const float* __restrict__ S, const float* __restrict__ Q,
    const float* __restrict__ rmax, const float* __restrict__ rsum,
    float* __restrict__ A) {
    const int b     = blockIdx.x;
    const int wave  = threadIdx.x >> 5;
    const int lane  = threadIdx.x & 31;
    const int row   = lane & 15;
    const int half  = lane >> 4;
    const int khalf = half << 1;
    const int nb    = blockIdx.y * 128 + (wave >> 1) * 32;
    const int db    = (wave & 1) * 64;

    const float* Qb    = Q + (long)b * Dd * Mm;
    const float* Srow0 = S + ((long)b * Nn + nb + row) * Mm;
    const float* Srow1 = Srow0 + 16L * Mm;
    const float  mx0   = rmax[b * Nn + nb + row];
    const float  in0   = 1.0f / rsum[b * Nn + nb + row];
    const float  mx1   = rmax[b * Nn + nb + 16 + row];
    const float  in1   = 1.0f / rsum[b * Nn + nb + 16 + row];

    v8f acc0[4] = {};
    v8f acc1[4] = {};
    for (int m0 = 0; m0 < Mm; m0 += 4) {
        const int kA = m0 + khalf;
        v2f a0, a1;
        a0.x = __expf(Srow0[kA] - mx0) * in0;
        a0.y = __expf(Srow0[kA + 1] - mx0) * in0;
        a1.x = __expf(Srow1[kA] - mx1) * in1;
        a1.y = __expf(Srow1[kA + 1] - mx1) * in1;
        #pragma unroll
        for (int dt = 0; dt < 4; ++dt) {
            const int d = db + dt * 16 + row;
            v2f bv;                           // B[k][col] = Qt[m0+k, d], contiguous pair
            bv.x = Qb[(long)d * Mm + kA];
            bv.y = Qb[(long)d * Mm + kA + 1];
            acc0[dt] = wmma4(a0, bv, acc0[dt]);
            acc1[dt] = wmma4(a1, bv, acc1[dt]);
        }
    }
    #pragma unroll
    for (int dt = 0; dt < 4; ++dt) {
        const int d = db + dt * 16 + row;
        #pragma unroll
        for (int r = 0; r < 8; ++r) {
            const int n0 = nb + r + half * 8;
            A[((long)b * Nn + n0) * Dd + d]      = acc0[dt][r];
            A[((long)b * Nn + n0 + 16) * Dd + d] = acc1[dt][r];
        }
    }
}

// T[b,m,d] = sum_n softmax_col(S)[n,m] * Ct[n,d]
// grid (B, M/128), block 256; wave: two 16-m strips x 64 d-cols, K=N=1024.
__global__ __launch_bounds__(256) void t_gemm_kernel(
    const float* __restrict__ S, const float* __restrict__ C,
    const float* __restrict__ cmax, const float* __restrict__ csum,
    float* __restrict__ T) {
    const int b     = blockIdx.x;
    const int wave  = threadIdx.x >> 5;
    const int lane  = threadIdx.x & 31;
    const int row   = lane & 15;
    const int half  = lane >> 4;
    const int khalf = half << 1;
    const int mb    = blockIdx.y * 128 + (wave >> 1) * 32;
    const int db    = (wave & 1) * 64;

    const float* Cb    = C + (long)b * Dd * Nn;
    const int    mA0   = mb + row;
    const int    mA1   = mA0 + 16;
    const float* Scol0 = S + (long)b * Nn * Mm + mA0;
    const float* Scol1 = Scol0 + 16;
    const float  mx0   = cmax[b * Mm + mA0];
    const float  in0   = 1.0f / csum[b * Mm + mA0];
    const float  mx1   = cmax[b * Mm + mA1];
    const float  in1   = 1.0f / csum[b * Mm + mA1];

    v8f acc0[4] = {};
    v8f acc1[4] = {};
    for (int n0 = 0; n0 < Nn; n0 += 4) {
        const int kA = n0 + khalf;
        v2f a0, a1;                           // A[m, n] = S_colnorm[n, m]
        a0.x = __expf(Scol0[(long)kA * Mm] - mx0) * in0;
        a0.y = __expf(Scol0[(long)(kA + 1) * Mm] - mx0) * in0;
        a1.x = __expf(Scol1[(long)kA * Mm] - mx1) * in1;
        a1.y = __expf(Scol1[(long)(kA + 1) * Mm] - mx1) * in1;
        #pragma unroll
        for (int dt = 0; dt < 4; ++dt) {
            const int d = db + dt * 16 + row;
            v2f bv;                           // B[k][col] = Ct[n0+k, d], contiguous pair
            bv.x = Cb[(long)d * Nn + kA];
            bv.y = Cb[(long)d * Nn + kA + 1];
            acc0[dt] = wmma4(a0, bv, acc0[dt]);
            acc1[dt] = wmma4(a1, bv, acc1[dt]);
        }
    }
    #pragma unroll
    for (int dt = 0; dt < 4; ++dt) {
        const int d = db + dt * 16 + row;
        #pragma unroll
        for (int r = 0; r < 8; ++r) {
            const int m0 = mb + r + half * 8;
            T[((long)b * Mm + m0) * Dd + d]      = acc0[dt][r];
            T[((long)b * Mm + m0 + 16) * Dd + d] = acc1[dt][r];
        }
    }
}

// Bt[b,n,d] = sum_m softmax_row(S)[n,m] * T[b,m,d]
// grid (B, N/128), block 256; same shape as a_gemm, B-operand from T (row-major).
__global__ __launch_bounds__(256) void bt_gemm_kernel(
    const float* __restrict__ S, const float* __restrict__ T,
    const float* __restrict__ rmax, const float* __restrict__ rsum,
    float* __restrict__ Bt) {
    const int b     = blockIdx.x;
    const int wave  = threadIdx.x >> 5;
    const int lane  = threadIdx.x & 31;
    const int row   = lane & 15;
    const int half  = lane >> 4;
    const int khalf = half << 1;
    const int nb    = blockIdx.y * 128 + (wave >> 1) * 32;
    const int db    = (wave & 1) * 64;

    const float* Tb    = T + (long)b * Mm * Dd;
    const float* Srow0 = S + ((long)b * Nn + nb + row) * Mm;
    const float* Srow1 = Srow0 + 16L * Mm;
    const float  mx0   = rmax[b * Nn + nb + row];
    const float  in0   = 1.0f / rsum[b * Nn + nb + row];
    const float  mx1   = rmax[b * Nn + nb + 16 + row];
    const float  in1   = 1.0f / rsum[b * Nn + nb + 16 + row];

    v8f acc0[4] = {};
    v8f acc1[4] = {};
    for (int m0 = 0; m0 < Mm; m0 += 4) {
        const int kA = m0 + khalf;
        v2f a0, a1;
        a0.x = __expf(Srow0[kA] - mx0) * in0;
        a0.y = __expf(Srow0[kA + 1] - mx0) * in0;
        a1.x = __expf(Srow1[kA] - mx1) * in1;
        a1.y = __expf(Srow1[kA + 1] - mx1) * in1;
        #pragma unroll
        for (int dt = 0; dt < 4; ++dt) {
            const int d = db + dt * 16 + row;
            v2f bv;                           // B[k][col] = T[m0+k, d]
            bv.x = Tb[(long)kA * Dd + d];
            bv.y = Tb[(long)(kA + 1) * Dd + d];
            acc0[dt] = wmma4(a0, bv, acc0[dt]);
            acc1[dt] = wmma4(a1, bv, acc1[dt]);
        }
    }
    #pragma unroll
    for (int dt = 0; dt < 4; ++dt) {
        const int d = db + dt * 16 + row;
        #pragma unroll
        for (int r = 0; r < 8; ++r) {
            const int n0 = nb + r + half * 8;
            Bt[((long)b * Nn + n0) * Dd + d]      = acc0[dt][r];
            Bt[((long)b * Nn + n0 + 16) * Dd + d] = acc1[dt][r];
        }
    }
}

// ---------------------------------------------------------------------------
extern "C" void kernel_launch(void* const* d_in, const int* in_sizes, int n_in,
                              void* d_out, int out_size, void* d_ws, size_t ws_size,
                              hipStream_t stream) {
    const float* C  = (const float*)d_in[0];
    const float* Q  = (const float*)d_in[1];
    const float* W0 = (const float*)d_in[2];
    const float* b0 = (const float*)d_in[3];
    float* out = (float*)d_out;

    // Workspace layout (floats):
    // S | rmax | rsum | cmax | csum | rterm | cterm | T | pmax | psum
    float* ws    = (float*)d_ws;
    float* S     = ws;                               // B*N*M = 16,777,216
    float* rmax  = S     + (long)Bsz * Nn * Mm;      // B*N
    float* rsum  = rmax  + (long)Bsz * Nn;           // B*N
    float* cmax  = rsum  + (long)Bsz * Nn;           // B*M
    float* csum  = cmax  + (long)Bsz * Mm;           // B*M
    float* rterm = csum  + (long)Bsz * Mm;           // B*N
    float* cterm = rterm + (long)Bsz * Nn;           // B*M
    float* T     = cterm + (long)Bsz * Mm;           // B*M*D = 2,097,152
    float* pmax  = T     + (long)Bsz * Mm * Dd;      // B*8*M
    float* psum  = pmax  + (long)Bsz * 8 * Mm;       // B*8*M

    float* Aout  = out;                              // B*N*D
    float* Btout = out + (long)Bsz * Nn * Dd;        // B*N*D

    rowterm_kernel  <<<dim3(Bsz, Nn / 256), 256, 0, stream>>>(C, W0, rterm);
    colterm_kernel  <<<dim3(Bsz),           256, 0, stream>>>(Q, W0, cterm);
    s_gemm_kernel   <<<dim3(Bsz, Nn / 128), 256, 0, stream>>>(C, Q, W0, b0, rterm, cterm, S);
    rowstats_kernel <<<dim3(Bsz, Nn / 8),   256, 0, stream>>>(S, rmax, rsum);
    colstats1_kernel<<<dim3(Bsz, Nn / 128), 256, 0, stream>>>(S, pmax, psum);
    colstats2_kernel<<<dim3(Bsz),           256, 0, stream>>>(pmax, psum, cmax, csum);
    a_gemm_kernel   <<<dim3(Bsz, Nn / 128), 256, 0, stream>>>(S, Q, rmax, rsum, Aout);
    t_gemm_kernel   <<<dim3(Bsz, Mm / 128), 256, 0, stream>>>(S, C, cmax, csum, T);
    bt_gemm_kernel  <<<dim3(Bsz, Nn / 128), 256, 0, stream>>>(S, T, rmax, rsum, Btout);
}